// MultiHeadedHockeyGNN_62242666054381
// MI455X (gfx1250) — compile-verified
//
#include <hip/hip_runtime.h>

#define BN_S 0.99999500003749975f   // 1/sqrt(1+1e-5)

typedef __attribute__((ext_vector_type(2))) float v2f;
typedef __attribute__((ext_vector_type(8))) float v8f;

enum { FLAG_RELU = 1, FLAG_BN = 2, FLAG_AAFF = 4, FLAG_BIAS = 8 };

// ---------------------------------------------------------------------------
// Generic f32 WMMA GEMM:  out[M,Ncols] = epilogue( A'[M,K] @ W[K,Ncols] )
//   A' rows optionally gathered (gidx) and optionally column-affine-BN'd.
//   Block = 256 threads = 8 waves; block tile = 16 rows x 128 cols,
//   each wave owns one 16x16 output tile, K-loop of V_WMMA_F32_16X16X4_F32.
//   M, Ncols multiples of 16; K a multiple of 4 (true for all uses here).
// ---------------------------------------------------------------------------
__global__ __launch_bounds__(256) void wmma_gemm_f32(
    const float* __restrict__ A, int lda, const int* __restrict__ gidx,
    const float* __restrict__ W, int ldw,
    const float* __restrict__ ag, const float* __restrict__ ab,
    const float* __restrict__ bias,
    const float* __restrict__ bng, const float* __restrict__ bnb,
    float* __restrict__ out, int ldo,
    int Ncols, int K, int flags)
{
    __shared__ float As[16][130];          // padded stride -> bank friendly
    const int tid  = threadIdx.x;
    const int wave = tid >> 5;
    const int lane = tid & 31;
    const int mBase = blockIdx.x * 16;

    // stage the 16 x K A tile in LDS (shared by all 8 waves)
    for (int idx = tid; idx < 16 * K; idx += 256) {
        int r = idx / K, c = idx - r * K;
        int row  = mBase + r;
        int grow = gidx ? gidx[row] : row;
        float v = A[(size_t)grow * lda + c];
        if (flags & FLAG_AAFF) v = ag[c] * (v * BN_S) + ab[c];
        As[r][c] = v;
    }
    __syncthreads();

    const int ct = blockIdx.y * 8 + wave;  // wave-uniform -> EXEC stays full
    if (ct * 16 >= Ncols) return;
    const int nBase = ct * 16;
    const int hi  = lane >> 4;             // lane half selects the K pair
    const int l15 = lane & 15;

    v8f acc = {};
    for (int kk = 0; kk < K; kk += 4) {
        const int ka = kk + 2 * hi;
        v2f a, b;
        a.x = As[l15][ka];
        a.y = As[l15][ka + 1];
        const float* wp = W + (size_t)ka * ldw + nBase + l15;
        b.x = wp[0];
        b.y = wp[ldw];
        acc = __builtin_amdgcn_wmma_f32_16x16x4_f32(
            false, a, false, b, (short)0, acc, false, false);
    }

    // epilogue: VGPR r holds rows r (lanes 0-15) and r+8 (lanes 16-31)
#pragma unroll
    for (int r = 0; r < 8; ++r) {
        int row = mBase + r + 8 * hi;
        int col = nBase + l15;
        float z = acc[r];
        if (flags & FLAG_BIAS) z += bias[col];
        if (flags & FLAG_BN)   z = bng[col] * (z * BN_S) + bnb[col];
        if (flags & FLAG_RELU) z = fmaxf(z, 0.0f);
        out[(size_t)row * ldo + col] = z;
    }
}

// ---------------------------------------------------------------------------
// CSR construction (amortized over the 3 GCN layers)
// ---------------------------------------------------------------------------
__global__ void fill_i32(int* __restrict__ p, int v, int n)
{
    int i = blockIdx.x * blockDim.x + threadIdx.x;
    if (i < n) p[i] = v;
}

__global__ void deg_count(const int* __restrict__ edst, int* __restrict__ deg, int E)
{
    int e = blockIdx.x * blockDim.x + threadIdx.x;
    if (e < E) atomicAdd(&deg[edst[e]], 1);
}

__global__ void make_dis(const int* __restrict__ deg, float* __restrict__ dis, int N)
{
    int i = blockIdx.x * blockDim.x + threadIdx.x;
    if (i < N) dis[i] = rsqrtf(1.0f + (float)deg[i]);  // self-loop adds 1
}

// single-workgroup exclusive scan over N counts -> row_ptr[N+1], cursor copy
__global__ __launch_bounds__(1024) void exclusive_scan(
    const int* __restrict__ deg, int* __restrict__ row_ptr,
    int* __restrict__ cursor, int N)
{
    __shared__ int tmp[1024];
    __shared__ int carry;
    const int t = threadIdx.x;
    if (t == 0) carry = 0;
    __syncthreads();
    for (int base = 0; base < N; base += 1024) {
        int x = (base + t < N) ? deg[base + t] : 0;
        int runbase = carry;
        tmp[t] = x;
        __syncthreads();
        for (int off = 1; off < 1024; off <<= 1) {
            int y = (t >= off) ? tmp[t - off] : 0;
            __syncthreads();
            tmp[t] += y;
            __syncthreads();
        }
        int incl = tmp[t];
        if (base + t < N) {
            int excl = runbase + incl - x;
            row_ptr[base + t] = excl;
            cursor[base + t]  = excl;
        }
        __syncthreads();
        if (t == 1023) carry = runbase + incl;
        __syncthreads();
    }
    if (t == 0) row_ptr[N] = carry;
}

__global__ void edge_scatter(const int* __restrict__ esrc, const int* __restrict__ edst,
                             int* __restrict__ cursor, int* __restrict__ csr_src, int E)
{
    int e = blockIdx.x * blockDim.x + threadIdx.x;
    if (e >= E) return;
    int pos = atomicAdd(&cursor[edst[e]], 1);
    csr_src[pos] = esrc[e];
}

// ---------------------------------------------------------------------------
// Fused GCN aggregation + self-loop + bias + BN + ReLU + residual.
// One wave32 per destination node; lane owns a float4 slice of the 128-wide
// row (32 x 4 = 128). Inner loop: broadcast csr_src/dis load + one coalesced
// 512B row read + FMA -> zero float atomics, no agg buffer, no extra pass.
// ---------------------------------------------------------------------------
__global__ __launch_bounds__(256) void gcn_gather_combine(
    const int* __restrict__ row_ptr, const int* __restrict__ csr_src,
    const float* __restrict__ dis, const float* __restrict__ hw,
    const float* __restrict__ cb,
    const float* __restrict__ bng, const float* __restrict__ bnb,
    float* __restrict__ h, int N)
{
    int wid = (int)(((size_t)blockIdx.x * blockDim.x + threadIdx.x) >> 5);
    if (wid >= N) return;                 // wave-uniform
    const int lane = threadIdx.x & 31;
    const int beg = row_ptr[wid];
    const int end = row_ptr[wid + 1];
    const float dn = dis[wid];

    float4 acc = make_float4(0.f, 0.f, 0.f, 0.f);
    for (int e = beg; e < end; ++e) {
        int s = csr_src[e];               // same address all lanes -> broadcast
        float c = dis[s];
        float4 v = ((const float4*)(hw + (size_t)s * 128))[lane];
        acc.x = fmaf(v.x, c, acc.x);
        acc.y = fmaf(v.y, c, acc.y);
        acc.z = fmaf(v.z, c, acc.z);
        acc.w = fmaf(v.w, c, acc.w);
    }

    const size_t base = (size_t)wid * 128 + lane * 4;
    float4 hv = *(const float4*)(hw + base);       // self-loop term source
    float4 ho = *(const float4*)(h + base);        // residual input
    const int c0 = lane * 4;
#pragma unroll
    for (int j = 0; j < 4; ++j) {
        float a = (&acc.x)[j];
        float s = (&hv.x)[j];
        // z = coef-agg * dn + hw * dn^2 + bias  ==  dn*(a + s*dn) + b
        float z = fmaf(dn, fmaf(s, dn, a), cb[c0 + j]);
        z = fmaf(bng[c0 + j], z * BN_S, bnb[c0 + j]);
        (&ho.x)[j] += fmaxf(z, 0.0f);
    }
    *(float4*)(h + base) = ho;
}

// ---------------------------------------------------------------------------
// concat 4 regression heads + outcome head: Wcat[128][320] + fused vectors
__global__ void pack_heads(const float* __restrict__ hW1, const float* __restrict__ hb1,
                           const float* __restrict__ hbng, const float* __restrict__ hbnb,
                           const float* __restrict__ oW1, const float* __restrict__ ob1,
                           const float* __restrict__ obng, const float* __restrict__ obnb,
                           float* __restrict__ Wcat, float* __restrict__ bcat,
                           float* __restrict__ gcat, float* __restrict__ bbcat)
{
    int idx = blockIdx.x * blockDim.x + threadIdx.x;
    if (idx >= 128 * 320) return;
    int k = idx / 320, c = idx - k * 320;
    float w;
    if (c < 256) { int hh = c >> 6, n = c & 63; w = hW1[hh * 128 * 64 + k * 64 + n]; }
    else         { w = oW1[k * 64 + (c - 256)]; }
    Wcat[k * 320 + c] = w;
    if (k == 0) {
        if (c < 256) { bcat[c] = hb1[c]; gcat[c] = hbng[c]; bbcat[c] = hbnb[c]; }
        else { int n = c - 256; bcat[c] = ob1[n]; gcat[c] = obng[n]; bbcat[c] = obnb[n]; }
    }
}

// last tiny layers: 4 dot(32) heads + [32x3] outcome matvec -> out[G,7]
__global__ void finalize_heads(const float* __restrict__ z2,
                               const float* __restrict__ hW3, const float* __restrict__ hb3,
                               const float* __restrict__ oW3, const float* __restrict__ ob3,
                               float* __restrict__ out, int G)
{
    int g = blockIdx.x * blockDim.x + threadIdx.x;
    if (g >= G) return;
    const float* z = z2 + (size_t)g * 160;
#pragma unroll
    for (int hh = 0; hh < 4; ++hh) {
        float s = hb3[hh];
#pragma unroll
        for (int k = 0; k < 32; ++k) s += z[hh * 32 + k] * hW3[hh * 32 + k];
        out[(size_t)g * 7 + hh] = s;
    }
#pragma unroll
    for (int j = 0; j < 3; ++j) {
        float s = ob3[j];
#pragma unroll
        for (int k = 0; k < 32; ++k) s += z[128 + k] * oW3[k * 3 + j];
        out[(size_t)g * 7 + 4 + j] = s;
    }
}

// ---------------------------------------------------------------------------
extern "C" void kernel_launch(void* const* d_in, const int* in_sizes, int n_in,
                              void* d_out, int out_size, void* d_ws, size_t ws_size,
                              hipStream_t stream)
{
    const float* x      = (const float*)d_in[0];
    const int*   esrc   = (const int*)d_in[1];
    const int*   edst   = (const int*)d_in[2];
    const int*   gidx   = (const int*)d_in[3];
    const float* in_g   = (const float*)d_in[4];
    const float* in_b   = (const float*)d_in[5];
    const float* lin_W  = (const float*)d_in[6];
    const float* lin_b  = (const float*)d_in[7];
    const float* bn0g   = (const float*)d_in[8];
    const float* bn0b   = (const float*)d_in[9];
    const float* cW[3]  = {(const float*)d_in[10], (const float*)d_in[14], (const float*)d_in[18]};
    const float* cb[3]  = {(const float*)d_in[11], (const float*)d_in[15], (const float*)d_in[19]};
    const float* bng[3] = {(const float*)d_in[12], (const float*)d_in[16], (const float*)d_in[20]};
    const float* bnb[3] = {(const float*)d_in[13], (const float*)d_in[17], (const float*)d_in[21]};
    const float* hW1 = (const float*)d_in[22];
    const float* hb1 = (const float*)d_in[23];
    const float* hbng = (const float*)d_in[24];
    const float* hbnb = (const float*)d_in[25];
    const float* hW2 = (const float*)d_in[26];
    const float* hb2 = (const float*)d_in[27];
    const float* hW3 = (const float*)d_in[28];
    const float* hb3 = (const float*)d_in[29];
    const float* oW1 = (const float*)d_in[30];
    const float* ob1 = (const float*)d_in[31];
    const float* obn1g = (const float*)d_in[32];
    const float* obn1b = (const float*)d_in[33];
    const float* oW2 = (const float*)d_in[34];
    const float* ob2 = (const float*)d_in[35];
    const float* obn2g = (const float*)d_in[36];
    const float* obn2b = (const float*)d_in[37];
    const float* oW3 = (const float*)d_in[38];
    const float* ob3 = (const float*)d_in[39];

    const int N = in_sizes[0] / 128;   // 50000
    const int E = in_sizes[1];         // 640000
    const int G = in_sizes[3];         // 16384
    const size_t NF = (size_t)N * 128;

    // workspace carve-up
    float* h    = (float*)d_ws;
    float* hw   = h    + NF;
    float* dis  = hw   + NF;
    float* z1   = dis  + N;
    float* z2   = z1   + (size_t)G * 320;
    float* Wcat = z2   + (size_t)G * 160;
    float* bcat = Wcat + 128 * 320;
    float* gcat = bcat + 320;
    float* bbc  = gcat + 320;
    int* deg     = (int*)(bbc + 320);
    int* row_ptr = deg + N;
    int* cursor  = row_ptr + (N + 1);
    int* csr_src = cursor + N;

    float* out = (float*)d_out;

    const int mBlocksN = N / 16;   // 3125
    const int mBlocksG = G / 16;   // 1024

    auto gemm = [&](const float* A, int lda, const int* gi,
                    const float* W, int ldw,
                    const float* ag, const float* ab,
                    const float* bias, const float* bg, const float* bb,
                    float* o, int ldo, int mBlocks, int Ncols, int K, int flags) {
        dim3 grid(mBlocks, (Ncols / 16 + 7) / 8);
        wmma_gemm_f32<<<grid, 256, 0, stream>>>(A, lda, gi, W, ldw, ag, ab,
                                                bias, bg, bb, o, ldo, Ncols, K, flags);
    };

    // 1) degree histogram -> dis = rsqrt(1+indeg), CSR by destination
    fill_i32<<<(N + 255) / 256, 256, 0, stream>>>(deg, 0, N);
    deg_count<<<(E + 255) / 256, 256, 0, stream>>>(edst, deg, E);
    make_dis<<<(N + 255) / 256, 256, 0, stream>>>(deg, dis, N);
    exclusive_scan<<<1, 1024, 0, stream>>>(deg, row_ptr, cursor, N);
    edge_scatter<<<(E + 255) / 256, 256, 0, stream>>>(esrc, edst, cursor, csr_src, E);

    // 2) h = relu(bn0( bn_in(x) @ lin_W + lin_b ))
    gemm(x, 128, nullptr, lin_W, 128, in_g, in_b, lin_b, bn0g, bn0b,
         h, 128, mBlocksN, 128, 128, FLAG_AAFF | FLAG_BIAS | FLAG_BN | FLAG_RELU);

    // 3) three residual GCN layers: WMMA GEMM + fused gather/combine (no atomics)
    for (int L = 0; L < 3; ++L) {
        gemm(h, 128, nullptr, cW[L], 128, nullptr, nullptr, nullptr, nullptr, nullptr,
             hw, 128, mBlocksN, 128, 128, 0);
        gcn_gather_combine<<<(int)(((size_t)N * 32 + 255) / 256), 256, 0, stream>>>(
            row_ptr, csr_src, dis, hw, cb[L], bng[L], bnb[L], h, N);
    }

    // 4) heads layer 1 (fused 4 heads + outcome): z1 = relu(bn(h[g] @ Wcat + bcat))
    pack_heads<<<(128 * 320 + 255) / 256, 256, 0, stream>>>(
        hW1, hb1, hbng, hbnb, oW1, ob1, obn1g, obn1b, Wcat, bcat, gcat, bbc);
    gemm(h, 128, gidx, Wcat, 320, nullptr, nullptr, bcat, gcat, bbc,
         z1, 320, mBlocksG, 320, 128, FLAG_BIAS | FLAG_BN | FLAG_RELU);

    // 5) heads layer 2 (block-diagonal): 4 regression heads (no BN) + outcome (BN)
    for (int hh = 0; hh < 4; ++hh) {
        gemm(z1 + hh * 64, 320, nullptr, hW2 + (size_t)hh * 64 * 32, 32,
             nullptr, nullptr, hb2 + hh * 32, nullptr, nullptr,
             z2 + hh * 32, 160, mBlocksG, 32, 64, FLAG_BIAS | FLAG_RELU);
    }
    gemm(z1 + 256, 320, nullptr, oW2, 32, nullptr, nullptr, ob2, obn2g, obn2b,
         z2 + 128, 160, mBlocksG, 32, 64, FLAG_BIAS | FLAG_BN | FLAG_RELU);

    // 6) final tiny layers -> out[G,7]
    finalize_heads<<<(G + 255) / 256, 256, 0, stream>>>(z2, hW3, hb3, oW3, ob3, out, G);
}